// GATNet_35055523070302
// MI455X (gfx1250) — compile-verified
//
#include <hip/hip_runtime.h>
#include <stdint.h>

#define NEG_SLOPE 0.2f

typedef float v2f __attribute__((ext_vector_type(2)));
typedef float v8f __attribute__((ext_vector_type(8)));

#if defined(__gfx1250__)
#if __has_builtin(__builtin_amdgcn_wmma_f32_16x16x4_f32)
#define HAVE_WMMA_F32X4 1
#endif
#if __has_builtin(__builtin_amdgcn_global_load_async_to_lds_b32)
#define HAVE_ASYNC_LDS 1
typedef __attribute__((address_space(1))) int as1_i32;   // param type per clang diag
typedef __attribute__((address_space(3))) int as3_i32;
#endif
#endif

__device__ __forceinline__ void wait_async_then_barrier() {
#if defined(HAVE_ASYNC_LDS)
#if __has_builtin(__builtin_amdgcn_s_wait_asynccnt)
    __builtin_amdgcn_s_wait_asynccnt(0);
#else
    asm volatile("s_wait_asynccnt 0x0" ::: "memory");
#endif
#endif
    __syncthreads();
}

// ---------------------------------------------------------------------------
// sortable-uint encoding of float -> u32 atomicMax works with memset(0) init.
__device__ __forceinline__ unsigned f2sort(float f) {
    unsigned b = __float_as_uint(f);
    return (b & 0x80000000u) ? ~b : (b | 0x80000000u);
}
__device__ __forceinline__ float sort2f(unsigned u) {
    unsigned b = (u & 0x80000000u) ? (u & 0x7fffffffu) : ~u;
    return __uint_as_float(b);
}

// ---------------------------------------------------------------------------
__global__ void edge_mean_kernel(const float* __restrict__ ea,
                                 float* __restrict__ emean, int E) {
    __shared__ float r0[256], r1[256], r2[256];
    float a0 = 0.f, a1 = 0.f, a2 = 0.f;
    for (int e = threadIdx.x; e < E; e += 256) {
        a0 += ea[3 * e + 0];
        a1 += ea[3 * e + 1];
        a2 += ea[3 * e + 2];
    }
    r0[threadIdx.x] = a0; r1[threadIdx.x] = a1; r2[threadIdx.x] = a2;
    __syncthreads();
    for (int s = 128; s > 0; s >>= 1) {
        if (threadIdx.x < (unsigned)s) {
            r0[threadIdx.x] += r0[threadIdx.x + s];
            r1[threadIdx.x] += r1[threadIdx.x + s];
            r2[threadIdx.x] += r2[threadIdx.x + s];
        }
        __syncthreads();
    }
    if (threadIdx.x == 0) {
        float inv = 1.f / (float)E;
        emean[0] = r0[0] * inv;
        emean[1] = r1[0] * inv;
        emean[2] = r2[0] * inv;
    }
}

// ---------------------------------------------------------------------------
// FAST GEMM: requires K % 4 == 0, Kout % 16 == 0, M % 16 == 0 (the two big
// layers). 256 threads = 8 waves; one N-tile per block, 8 M-tiles (one per
// wave). B panel (K x 16) staged once into LDS in (k-pair, col) order so a
// lane's B fragment is a single ds_load_b64; A fragment is a global b64.
__global__ void gemm_bias_wmma_fast(const float* __restrict__ X,
                                    const float* __restrict__ W,
                                    const float* __restrict__ bias,
                                    float* __restrict__ Y,
                                    int M, int K, int Kout, int tilesM) {
    __shared__ float ldsB[512 * 16];             // 32 KB max (K<=512)
    const int tn = blockIdx.y * 16;

    // stage B panel: W[k][tn+col] -> ldsB[(k>>1)*32 + col*2 + (k&1)]
    for (int idx = threadIdx.x; idx < K * 16; idx += 256) {
        int k = idx >> 4, col = idx & 15;
        int loff = ((k >> 1) << 5) + (col << 1) + (k & 1);
        const float* gp = W + (size_t)k * Kout + tn + col;
#if defined(HAVE_ASYNC_LDS)
        __builtin_amdgcn_global_load_async_to_lds_b32(
            (as1_i32*)(uintptr_t)gp,
            (as3_i32*)(unsigned)(uintptr_t)(ldsB + loff), 0, 0);
#else
        ldsB[loff] = *gp;
#endif
    }
    wait_async_then_barrier();

    const int wv   = threadIdx.x >> 5;
    const int lane = threadIdx.x & 31;
    const int hf   = lane >> 4;
    const int l15  = lane & 15;
    const int tileM = blockIdx.x * 8 + wv;
    if (tileM >= tilesM) return;                 // wave-uniform

    const int tm = tileM * 16;
    const float* __restrict__ Xr = X + (size_t)(tm + l15) * K;
    const float* __restrict__ Bl = ldsB + (size_t)l15 * 2;

#if defined(HAVE_WMMA_F32X4)
    v8f acc = {0.f, 0.f, 0.f, 0.f, 0.f, 0.f, 0.f, 0.f};
#pragma unroll 4
    for (int k0 = 0; k0 < K; k0 += 4) {
        v2f a = *(const v2f*)(Xr + k0 + hf * 2);                 // global b64
        v2f b = *(const v2f*)(Bl + (((k0 >> 1) + hf) << 5));     // ds b64
        acc = __builtin_amdgcn_wmma_f32_16x16x4_f32(false, a, false, b,
                                                    (short)0, acc, false, false);
    }
    {
        float bv = bias[tn + l15];
#pragma unroll
        for (int r = 0; r < 8; ++r)
            Y[(size_t)(tm + r + hf * 8) * Kout + tn + l15] = acc[r] + bv;
    }
#else
    {
        float accs[8] = {0.f, 0.f, 0.f, 0.f, 0.f, 0.f, 0.f, 0.f};
        for (int k = 0; k < K; ++k) {
            float wvv = ldsB[((k >> 1) << 5) + (l15 << 1) + (k & 1)];
#pragma unroll
            for (int r = 0; r < 8; ++r)
                accs[r] += X[(size_t)(tm + r + hf * 8) * K + k] * wvv;
        }
        float bv = bias[tn + l15];
#pragma unroll
        for (int r = 0; r < 8; ++r)
            Y[(size_t)(tm + r + hf * 8) * Kout + tn + l15] = accs[r] + bv;
    }
#endif
}

// ---------------------------------------------------------------------------
// GENERIC GEMM for the ragged layers (K=2 or Kout=4). Branch-free inner loop:
// clamp addresses into bounds (unconditional loads) and zero-select values
// (v_cndmask), so EXEC stays untouched before the WMMAs.
__global__ void gemm_bias_wmma_gen(const float* __restrict__ X,
                                   const float* __restrict__ W,
                                   const float* __restrict__ bias,
                                   float* __restrict__ Y,
                                   int M, int K, int Kout, int tilesN,
                                   int numTiles) {
    int tile = blockIdx.x * 4 + (threadIdx.x >> 5);
    if (tile >= numTiles) return;                // wave-uniform
    const int lane = threadIdx.x & 31;
    const int hf   = lane >> 4;
    const int l15  = lane & 15;
    const int tm = (tile / tilesN) * 16;
    const int tn = (tile % tilesN) * 16;
    const int row = tm + l15;
    const int col = tn + l15;
    const bool cok = (col < Kout);
    const int colS = cok ? col : tn;             // tn always < Kout

#if defined(HAVE_WMMA_F32X4)
    v8f acc = {0.f, 0.f, 0.f, 0.f, 0.f, 0.f, 0.f, 0.f};
    for (int k0 = 0; k0 < K; k0 += 4) {
        int ka  = k0 + hf * 2;
        int ka0 = (ka < K) ? ka : 0;
        int ka1 = (ka + 1 < K) ? ka + 1 : 0;
        float ax = X[(size_t)row * K + ka0];
        float ay = X[(size_t)row * K + ka1];
        float bx = W[(size_t)ka0 * Kout + colS];
        float by = W[(size_t)ka1 * Kout + colS];
        v2f a, b;
        a.x = (ka     < K) ? ax : 0.f;
        a.y = (ka + 1 < K) ? ay : 0.f;
        b.x = (cok && ka     < K) ? bx : 0.f;
        b.y = (cok && ka + 1 < K) ? by : 0.f;
        acc = __builtin_amdgcn_wmma_f32_16x16x4_f32(false, a, false, b,
                                                    (short)0, acc, false, false);
    }
    if (cok) {
        float bv = bias[col];
#pragma unroll
        for (int r = 0; r < 8; ++r)
            Y[(size_t)(tm + r + hf * 8) * Kout + col] = acc[r] + bv;
    }
#else
    if (cok) {
        float accs[8] = {0.f, 0.f, 0.f, 0.f, 0.f, 0.f, 0.f, 0.f};
        for (int k = 0; k < K; ++k) {
            float wvv = W[(size_t)k * Kout + col];
#pragma unroll
            for (int r = 0; r < 8; ++r)
                accs[r] += X[(size_t)(tm + r + hf * 8) * K + k] * wvv;
        }
        float bv = bias[col];
#pragma unroll
        for (int r = 0; r < 8; ++r)
            Y[(size_t)(tm + r + hf * 8) * Kout + col] = accs[r] + bv;
    }
#endif
}

// ---------------------------------------------------------------------------
// Per-edge logits, float4-vectorized: 4 channels/lane/iter, one ds_add_f32
// per 4 contributions; segment-max via sortable-u32 atomicMax.
__global__ void edge_logits_kernel(const float* __restrict__ xl,
                                   const float* __restrict__ xr,
                                   const long long* __restrict__ srcp,
                                   const long long* __restrict__ dstp,
                                   const float* __restrict__ eattr,
                                   const float* __restrict__ emean,
                                   const float* __restrict__ We,
                                   const float* __restrict__ att,
                                   float* __restrict__ logits,
                                   unsigned* __restrict__ mxu,
                                   int E, int Esl, int H, int HC, int cshift) {
    int wv = threadIdx.x >> 5, lane = threadIdx.x & 31;
    long long e = (long long)blockIdx.x * 4 + wv;
    bool valid = (e < Esl);
    long long s = 0, d = 0;
    float e0 = 0.f, e1 = 0.f, e2 = 0.f;
    if (valid) {
        if (e < E) {
            s = srcp[e]; d = dstp[e];
            e0 = eattr[e * 3 + 0]; e1 = eattr[e * 3 + 1]; e2 = eattr[e * 3 + 2];
        } else {
            s = e - E; d = e - E;
            e0 = emean[0]; e1 = emean[1]; e2 = emean[2];
        }
    }
    __shared__ float hs[4 * 8];
    if (lane < 8) hs[wv * 8 + lane] = 0.f;
    __syncthreads();
    if (valid) {
        const float* xls = xl + (size_t)s * HC;
        const float* xrd = xr + (size_t)d * HC;
        __builtin_prefetch(xls, 0, 0);
        __builtin_prefetch(xrd, 0, 0);
        for (int i = lane * 4; i < HC; i += 128) {
            float4 w0 = *(const float4*)(We + i);
            float4 w1 = *(const float4*)(We + HC + i);
            float4 w2 = *(const float4*)(We + 2 * HC + i);
            float4 xa = *(const float4*)(xls + i);
            float4 xb = *(const float4*)(xrd + i);
            float4 at = *(const float4*)(att + i);
            float m0 = xa.x + xb.x + e0 * w0.x + e1 * w1.x + e2 * w2.x;
            float m1 = xa.y + xb.y + e0 * w0.y + e1 * w1.y + e2 * w2.y;
            float m2 = xa.z + xb.z + e0 * w0.z + e1 * w1.z + e2 * w2.z;
            float m3 = xa.w + xb.w + e0 * w0.w + e1 * w1.w + e2 * w2.w;
            float c = at.x * ((m0 > 0.f) ? m0 : NEG_SLOPE * m0)
                    + at.y * ((m1 > 0.f) ? m1 : NEG_SLOPE * m1)
                    + at.z * ((m2 > 0.f) ? m2 : NEG_SLOPE * m2)
                    + at.w * ((m3 > 0.f) ? m3 : NEG_SLOPE * m3);
            atomicAdd(&hs[wv * 8 + (i >> cshift)], c);       // ds_add_f32
        }
    }
    __syncthreads();
    if (valid && lane < H) {
        float lg = hs[wv * 8 + lane];
        logits[e * H + lane] = lg;
        atomicMax(&mxu[d * H + lane], f2sort(lg));           // global u32 max
    }
}

// ---------------------------------------------------------------------------
__global__ void edge_exp_kernel(float* __restrict__ lg,
                                const unsigned* __restrict__ mxu,
                                float* __restrict__ den,
                                const long long* __restrict__ dstp,
                                int E, int Esl, int H) {
    long long idx = (long long)blockIdx.x * blockDim.x + threadIdx.x;
    if (idx >= (long long)Esl * H) return;
    long long e = idx / H;
    int hh = (int)(idx % H);
    long long d = (e < E) ? (long long)dstp[e] : (e - E);
    float mx = sort2f(mxu[d * H + hh]);
    float ex = __expf(lg[idx] - mx);
    lg[idx] = ex;
    atomicAdd(&den[d * H + hh], ex);                          // global_atomic_add_f32
}

// ---------------------------------------------------------------------------
__global__ void edge_scatter_kernel(float* __restrict__ lgex,
                                    const float* __restrict__ den,
                                    const float* __restrict__ xl,
                                    const long long* __restrict__ srcp,
                                    const long long* __restrict__ dstp,
                                    float* __restrict__ outb,
                                    int E, int Esl, int H, int HC, int cshift) {
    int wv = threadIdx.x >> 5, lane = threadIdx.x & 31;
    long long e = (long long)blockIdx.x * 4 + wv;
    bool valid = (e < Esl);
    long long s = 0, d = 0;
    if (valid) {
        if (e < E) { s = srcp[e]; d = dstp[e]; }
        else       { s = e - E;  d = e - E;   }
    }
    __shared__ float hal[4 * 8];
    if (valid && lane < H) {
        float a = lgex[e * H + lane] / den[d * H + lane];
        lgex[e * H + lane] = a;                 // alpha kept for output
        hal[wv * 8 + lane] = a;
    }
    __syncthreads();
    if (valid) {
        const float* xls = xl + (size_t)s * HC;
        float* od = outb + (size_t)d * HC;
        __builtin_prefetch(xls, 0, 0);
        for (int i = lane * 4; i < HC; i += 128) {
            float al = hal[wv * 8 + (i >> cshift)];
            float4 xa = *(const float4*)(xls + i);
            atomicAdd(&od[i + 0], al * xa.x);
            atomicAdd(&od[i + 1], al * xa.y);
            atomicAdd(&od[i + 2], al * xa.z);
            atomicAdd(&od[i + 3], al * xa.w);
        }
    }
}

// ---------------------------------------------------------------------------
__global__ void finalize_concat_kernel(float* __restrict__ buf,
                                       const float* __restrict__ bias,
                                       int tot, int HC) {
    int idx = blockIdx.x * blockDim.x + threadIdx.x;
    if (idx >= tot) return;
    float y = buf[idx] + bias[idx % HC];
    y = (y > 0.f) ? y : (__expf(y) - 1.f);      // ELU between layers
    buf[idx] = y;
}

__global__ void finalize_mean_kernel(const float* __restrict__ buf,
                                     const float* __restrict__ bias,
                                     float* __restrict__ dstbuf,
                                     int Nn, int H, int C) {
    int idx = blockIdx.x * blockDim.x + threadIdx.x;
    if (idx >= Nn * C) return;
    int n = idx / C, j = idx % C;
    float acc = 0.f;
    for (int hh = 0; hh < H; ++hh) acc += buf[(size_t)n * H * C + hh * C + j];
    dstbuf[idx] = acc / (float)H + bias[j];
}

__global__ void write_edges_alpha_kernel(const long long* __restrict__ srcp,
                                         const long long* __restrict__ dstp,
                                         const float* __restrict__ alpha,
                                         float* __restrict__ out,
                                         int E, int Esl) {
    int t = blockIdx.x * blockDim.x + threadIdx.x;
    if (t >= 3 * Esl) return;
    float v;
    if (t < Esl)           v = (t < E) ? (float)srcp[t] : (float)(t - E);
    else if (t < 2 * Esl) { int tt = t - Esl; v = (tt < E) ? (float)dstp[tt] : (float)(tt - E); }
    else                   v = alpha[t - 2 * Esl];
    out[t] = v;
}

// ---------------------------------------------------------------------------
extern "C" void kernel_launch(void* const* d_in, const int* in_sizes, int n_in,
                              void* d_out, int out_size, void* d_ws, size_t ws_size,
                              hipStream_t stream) {
    (void)n_in; (void)out_size; (void)ws_size;
    const float*     x     = (const float*)d_in[0];
    const long long* ei    = (const long long*)d_in[1];
    const float*     eattr = (const float*)d_in[2];
    const int Nn  = in_sizes[0] / 2;     // x is [N,2]
    const int E   = in_sizes[1] / 2;     // edge_index is [2,E]
    const int Esl = E + Nn;
    const long long* srcp = ei;
    const long long* dstp = ei + E;

    static const int cfg_fin[4] = {2, 128, 512, 128};
    static const int cfg_c[4]   = {16, 64, 32, 4};
    static const int cfg_h[4]   = {8, 8, 4, 1};

    uintptr_t w = (uintptr_t)d_ws;
    auto carve = [&](size_t bytes) -> void* {
        void* p = (void*)w;
        w += (bytes + 255) & ~(size_t)255;
        return p;
    };
    float*    emean = (float*)carve(64);
    float*    bufA  = (float*)carve((size_t)Nn * 512 * sizeof(float));
    float*    bufB  = (float*)carve((size_t)Nn * 512 * sizeof(float));
    float*    xlb   = (float*)carve((size_t)Nn * 512 * sizeof(float));
    float*    xrb   = (float*)carve((size_t)Nn * 512 * sizeof(float));
    float*    lgb   = (float*)carve((size_t)Esl * 8 * sizeof(float));
    float*    denb  = (float*)carve((size_t)Nn * 8 * sizeof(float));
    unsigned* mxu   = (unsigned*)carve((size_t)Nn * 8 * sizeof(unsigned));

    edge_mean_kernel<<<1, 256, 0, stream>>>(eattr, emean, E);

    const float* cur = x;
    float* outs[4] = {bufA, bufB, bufA, bufB};   // ping-pong; layer input never == outs[L]
    const int tilesM = Nn / 16;                  // 20000/16 = 1250

    for (int L = 0; L < 4; ++L) {
        const int pb = 3 + 7 * L;
        const float* Wl   = (const float*)d_in[pb + 0];
        const float* bl   = (const float*)d_in[pb + 1];
        const float* Wr   = (const float*)d_in[pb + 2];
        const float* br   = (const float*)d_in[pb + 3];
        const float* We   = (const float*)d_in[pb + 4];
        const float* att  = (const float*)d_in[pb + 5];
        const float* bias = (const float*)d_in[pb + 6];
        const int fin = cfg_fin[L], C = cfg_c[L], H = cfg_h[L];
        const int HC = H * C;
        const int cshift = __builtin_ctz((unsigned)C);

        if ((fin % 4 == 0) && (HC % 16 == 0)) {
            dim3 g((tilesM + 7) / 8, HC / 16);
            gemm_bias_wmma_fast<<<g, 256, 0, stream>>>(cur, Wl, bl, xlb, Nn, fin, HC, tilesM);
            gemm_bias_wmma_fast<<<g, 256, 0, stream>>>(cur, Wr, br, xrb, Nn, fin, HC, tilesM);
        } else {
            int tilesN   = (HC + 15) / 16;
            int numTiles = tilesM * tilesN;
            int gBlocks  = (numTiles + 3) / 4;
            gemm_bias_wmma_gen<<<gBlocks, 128, 0, stream>>>(cur, Wl, bl, xlb, Nn, fin, HC, tilesN, numTiles);
            gemm_bias_wmma_gen<<<gBlocks, 128, 0, stream>>>(cur, Wr, br, xrb, Nn, fin, HC, tilesN, numTiles);
        }

        (void)hipMemsetAsync(mxu,  0, (size_t)Nn * H * sizeof(unsigned), stream);
        (void)hipMemsetAsync(denb, 0, (size_t)Nn * H * sizeof(float), stream);
        float* outb = outs[L];
        (void)hipMemsetAsync(outb, 0, (size_t)Nn * HC * sizeof(float), stream);

        int eBlocks = (Esl + 3) / 4;
        edge_logits_kernel<<<eBlocks, 128, 0, stream>>>(xlb, xrb, srcp, dstp, eattr, emean,
                                                        We, att, lgb, mxu, E, Esl, H, HC, cshift);
        long long tot = (long long)Esl * H;
        edge_exp_kernel<<<(int)((tot + 255) / 256), 256, 0, stream>>>(lgb, mxu, denb, dstp, E, Esl, H);
        edge_scatter_kernel<<<eBlocks, 128, 0, stream>>>(lgb, denb, xlb, srcp, dstp, outb,
                                                         E, Esl, H, HC, cshift);

        if (L < 3) {
            int tot2 = Nn * HC;
            finalize_concat_kernel<<<(tot2 + 255) / 256, 256, 0, stream>>>(outb, bias, tot2, HC);
            cur = outb;
        } else {
            int tot2 = Nn * C;
            finalize_mean_kernel<<<(tot2 + 255) / 256, 256, 0, stream>>>(outb, bias, (float*)d_out, Nn, H, C);
        }
    }

    int tot3 = 3 * Esl;
    write_edges_alpha_kernel<<<(tot3 + 255) / 256, 256, 0, stream>>>(
        srcp, dstp, lgb, (float*)d_out + (size_t)Nn * 4, E, Esl);
}